// Tail_10256381903697
// MI455X (gfx1250) — compile-verified
//
#include <hip/hip_runtime.h>
#include <math.h>

typedef __attribute__((ext_vector_type(2))) float v2f;
typedef __attribute__((ext_vector_type(8))) float v8f;

#define NB   32
#define NH   32
#define NKV  8
#define NG   4
#define NS   4096
#define ND   64
#define NHID 2048
#define NFF  8192
#define NV   32000

__device__ __forceinline__ v8f wmma4(v2f a, v2f b, v8f c) {
  // D(16x16,f32) = A(16x4,f32) * B(4x16,f32) + C
  return __builtin_amdgcn_wmma_f32_16x16x4_f32(false, a, false, b, (short)0, c,
                                               false, false);
}

__device__ __forceinline__ float gelu_exact(float x) {
  return 0.5f * x * (1.0f + erff(x * 0.70710678118654752f));
}

// -------------------- GQA decode attention --------------------
// one workgroup per (b, kv); streams K,V once for all 4 grouped q heads.
__global__ __launch_bounds__(256) void attn_kernel(
    const float* __restrict__ q, const float* __restrict__ ks,
    const float* __restrict__ vs, float* __restrict__ ctx) {
  __shared__ float Ps[NG][NS];            // scores -> exp'd probs (64 KB)
  __shared__ float qs[NG][ND];
  __shared__ float rbuf[256];
  __shared__ float mx[NG];
  __shared__ float sm[NG];
  __shared__ float part[2][NG][NG][16];   // [s-half][d-tile][g][n]

  const int b    = blockIdx.x / NKV;
  const int kv   = blockIdx.x % NKV;
  const int tid  = threadIdx.x;
  const int wave = tid >> 5;
  const int lane = tid & 31;
  const int r    = lane & 15;            // M row (A) / N col (B)
  const int koff = (lane >> 4) * 2;      // K split across lane halves

  {   // stage q for the 4 grouped heads
    int g = tid >> 6, d = tid & 63;
    qs[g][d] = q[(size_t)(b * NH + kv * NG + g) * ND + d];
  }
  __syncthreads();

  const float* Kb = ks + (size_t)(b * NKV + kv) * NS * ND;
  const float* Vb = vs + (size_t)(b * NKV + kv) * NS * ND;

  // ---- phase 1: scores = (q @ K^T) * 1/sqrt(D); wave w owns s in [w*512, w*512+512)
  for (int t = 0; t < 32; ++t) {
    const int s0 = wave * 512 + t * 16;
    v8f c = {0, 0, 0, 0, 0, 0, 0, 0};
#pragma unroll 4
    for (int k0 = 0; k0 < ND; k0 += 4) {
      v2f a = *(const v2f*)&qs[r & 3][k0 + koff];  // rows 4..15 replicate 0..3
      v2f bb = __builtin_nontemporal_load(
          (const v2f*)(Kb + (size_t)(s0 + r) * ND + k0 + koff));
      c = wmma4(a, bb, c);
    }
    if (lane < 16) {
#pragma unroll
      for (int i = 0; i < 4; ++i) Ps[i][s0 + lane] = c[i] * 0.125f;
    }
  }
  __syncthreads();

  // ---- phase 2a: per-head max
  {
    const int g = tid >> 6, t = tid & 63;
    float m = -3.0e38f;
    for (int s = t; s < NS; s += 64) m = fmaxf(m, Ps[g][s]);
    rbuf[tid] = m;
    __syncthreads();
    for (int off = 32; off > 0; off >>= 1) {
      if (t < off) rbuf[tid] = fmaxf(rbuf[tid], rbuf[tid + off]);
      __syncthreads();
    }
    if (t == 0) mx[g] = rbuf[tid];
  }
  __syncthreads();
  // ---- phase 2b: exponentiate in place + per-head sum
  {
    const int g = tid >> 6, t = tid & 63;
    const float m = mx[g];
    float acc = 0.f;
    for (int s = t; s < NS; s += 64) {
      float p = __expf(Ps[g][s] - m);
      Ps[g][s] = p;
      acc += p;
    }
    rbuf[tid] = acc;
    __syncthreads();
    for (int off = 32; off > 0; off >>= 1) {
      if (t < off) rbuf[tid] += rbuf[tid + off];
      __syncthreads();
    }
    if (t == 0) sm[g] = rbuf[tid];
  }
  __syncthreads();

  // ---- phase 3: ctx = P @ V ; wave = (s-half, d-tile); V streamed once
  {
    const int dt = wave & 3, sh = wave >> 2;
    const int d0 = dt * 16;
    const int kbeg = sh * (NS / 2);
    v8f c = {0, 0, 0, 0, 0, 0, 0, 0};
#pragma unroll 2
    for (int k0 = kbeg; k0 < kbeg + NS / 2; k0 += 4) {
      v2f a = *(const v2f*)&Ps[r & 3][k0 + koff];
      v2f bb;
      bb.x = __builtin_nontemporal_load(Vb + (size_t)(k0 + koff) * ND + d0 + r);
      bb.y = __builtin_nontemporal_load(Vb + (size_t)(k0 + koff + 1) * ND + d0 + r);
      c = wmma4(a, bb, c);
    }
    if (lane < 16) {
#pragma unroll
      for (int i = 0; i < 4; ++i) part[sh][dt][i][lane] = c[i];
    }
  }
  __syncthreads();
  {   // combine halves, normalize, scatter to [b, (kv*4+g)*64 + d]
    const int g = tid >> 6, d = tid & 63;
    const int dt2 = d >> 4, dl = d & 15;
    float v = (part[0][dt2][g][dl] + part[1][dt2][g][dl]) / sm[g];
    ctx[(size_t)b * NHID + (size_t)(kv * NG + g) * ND + d] = v;
  }
}

// -------------------- fused GEMM (single K pass): out[32,N] = X @ W^T (+bias)(+gelu)(+add)
// used when N/128 alone gives enough blocks (LM head).
__global__ __launch_bounds__(256) void gemm_kernel(
    const float* __restrict__ X, const float* __restrict__ W,
    const float* __restrict__ bias, const float* __restrict__ addp,
    float* __restrict__ out, int K, int N, int act) {
  const int wave = threadIdx.x >> 5;
  const int lane = threadIdx.x & 31;
  const int r    = lane & 15;
  const int koff = (lane >> 4) * 2;
  const int n0   = blockIdx.x * 128 + wave * 16;

  const float* x0 = X + (size_t)r * K + koff;
  const float* x1 = X + (size_t)(r + 16) * K + koff;
  const float* wp = W + (size_t)(n0 + r) * K + koff;

  v8f c0 = {0, 0, 0, 0, 0, 0, 0, 0};
  v8f c1 = {0, 0, 0, 0, 0, 0, 0, 0};
#pragma unroll 4
  for (int k0 = 0; k0 < K; k0 += 4) {
    v2f a0 = *(const v2f*)(x0 + k0);
    v2f a1 = *(const v2f*)(x1 + k0);
    v2f bb = __builtin_nontemporal_load((const v2f*)(wp + k0));
    c0 = wmma4(a0, bb, c0);
    c1 = wmma4(a1, bb, c1);
  }
  const int half = lane >> 4;
  const int n = n0 + r;
  const float bv = bias ? bias[n] : 0.f;
#pragma unroll
  for (int i = 0; i < 8; ++i) {
    const int m = half * 8 + i;   // C layout: VGPR i -> rows i / i+8 by lane half
    float v0 = c0[i] + bv;
    float v1 = c1[i] + bv;
    if (act) { v0 = gelu_exact(v0); v1 = gelu_exact(v1); }
    if (addp) {
      v0 += addp[(size_t)m * N + n];
      v1 += addp[(size_t)(m + 16) * N + n];
    }
    out[(size_t)m * N + n] = v0;
    out[(size_t)(m + 16) * N + n] = v1;
  }
}

// -------------------- K-split GEMM partial: grid (N/128, KS) --------------------
// block (bx, ks) computes X[:, ksrange] @ W[n0:n0+128, ksrange]^T -> partial[ks][32][N]
// many concurrent weight streams => enough WGPs in flight to saturate HBM.
__global__ __launch_bounds__(256) void gemm_part_kernel(
    const float* __restrict__ X, const float* __restrict__ W,
    float* __restrict__ partial, int K, int N, int KS) {
  const int wave = threadIdx.x >> 5;
  const int lane = threadIdx.x & 31;
  const int r    = lane & 15;
  const int koff = (lane >> 4) * 2;
  const int n0   = blockIdx.x * 128 + wave * 16;
  const int ksid = blockIdx.y;
  const int Kc   = K / KS;
  const int kbeg = ksid * Kc;

  const float* x0 = X + (size_t)r * K + koff;
  const float* x1 = X + (size_t)(r + 16) * K + koff;
  const float* wp = W + (size_t)(n0 + r) * K + koff;

  v8f c0 = {0, 0, 0, 0, 0, 0, 0, 0};
  v8f c1 = {0, 0, 0, 0, 0, 0, 0, 0};
#pragma unroll 4
  for (int k0 = kbeg; k0 < kbeg + Kc; k0 += 4) {
    v2f a0 = *(const v2f*)(x0 + k0);
    v2f a1 = *(const v2f*)(x1 + k0);
    v2f bb = __builtin_nontemporal_load((const v2f*)(wp + k0));
    c0 = wmma4(a0, bb, c0);
    c1 = wmma4(a1, bb, c1);
  }
  const int half = lane >> 4;
  const int n = n0 + r;
  float* pb = partial + (size_t)ksid * 32 * N;
#pragma unroll
  for (int i = 0; i < 8; ++i) {
    const int m = half * 8 + i;
    pb[(size_t)m * N + n] = c0[i];
    pb[(size_t)(m + 16) * N + n] = c1[i];
  }
}

// -------------------- finalize: out = [act](sum_ks partial + bias) [+ add] ----
__global__ __launch_bounds__(256) void finalize_kernel(
    const float* __restrict__ partial, const float* __restrict__ bias,
    const float* __restrict__ addp, float* __restrict__ out, int N, int KS,
    int act) {
  const int idx = blockIdx.x * 256 + threadIdx.x;   // over 32*N
  const int n = idx % N;
  float v = 0.f;
  for (int ksid = 0; ksid < KS; ++ksid)             // fixed order: deterministic
    v += partial[(size_t)ksid * 32 * N + idx];
  if (bias) v += bias[n];
  if (act) v = gelu_exact(v);
  if (addp) v += addp[idx];
  out[idx] = v;
}

// -------------------- LayerNorm over last dim (N=2048), one row per block
__global__ __launch_bounds__(256) void ln_kernel(
    const float* __restrict__ X, const float* __restrict__ g,
    const float* __restrict__ bta, float* __restrict__ out, int N) {
  __shared__ float rb[256];
  const int row = blockIdx.x;
  const int t = threadIdx.x;
  const float* x = X + (size_t)row * N;

  float s = 0.f;
  for (int i = t; i < N; i += 256) s += x[i];
  rb[t] = s;
  __syncthreads();
  for (int off = 128; off > 0; off >>= 1) {
    if (t < off) rb[t] += rb[t + off];
    __syncthreads();
  }
  const float mean = rb[0] / (float)N;
  __syncthreads();

  float vacc = 0.f;
  for (int i = t; i < N; i += 256) {
    float d = x[i] - mean;
    vacc += d * d;
  }
  rb[t] = vacc;
  __syncthreads();
  for (int off = 128; off > 0; off >>= 1) {
    if (t < off) rb[t] += rb[t + off];
    __syncthreads();
  }
  const float inv = rsqrtf(rb[0] / (float)N + 1e-5f);

  for (int i = t; i < N; i += 256)
    out[(size_t)row * N + i] = (x[i] - mean) * inv * g[i] + bta[i];
}

extern "C" void kernel_launch(void* const* d_in, const int* in_sizes, int n_in,
                              void* d_out, int out_size, void* d_ws,
                              size_t ws_size, hipStream_t stream) {
  (void)in_sizes; (void)n_in; (void)out_size; (void)ws_size;
  const float* residual = (const float*)d_in[0];
  const float* q   = (const float*)d_in[1];
  const float* ks  = (const float*)d_in[2];
  const float* vs  = (const float*)d_in[3];
  const float* Wo  = (const float*)d_in[4];
  const float* bo  = (const float*)d_in[5];
  const float* g2  = (const float*)d_in[6];
  const float* b2  = (const float*)d_in[7];
  const float* W1  = (const float*)d_in[8];
  const float* b1  = (const float*)d_in[9];
  const float* W2  = (const float*)d_in[10];
  const float* b2m = (const float*)d_in[11];
  const float* gf  = (const float*)d_in[12];
  const float* bf  = (const float*)d_in[13];
  const float* Wlm = (const float*)d_in[14];
  float* out = (float*)d_out;

  float* ws  = (float*)d_ws;
  float* ctx  = ws;                 // 32*2048
  float* h    = ws + 65536;         // res2
  float* ln2  = ws + 131072;
  float* ffn  = ws + 196608;        // 32*8192
  float* h2   = ws + 458752;
  float* hf   = ws + 524288;
  float* part = ws + 589824;        // up to 4*32*8192 = 1M floats (4 MB)

  attn_kernel<<<NB * NKV, 256, 0, stream>>>(q, ks, vs, ctx);

  // h = ctx @ Wo^T + bo + residual  (K=2048 split 8 ways -> 128 blocks)
  gemm_part_kernel<<<dim3(NHID / 128, 8), 256, 0, stream>>>(ctx, Wo, part, NHID, NHID, 8);
  finalize_kernel<<<(NB * NHID) / 256, 256, 0, stream>>>(part, bo, residual, h, NHID, 8, 0);

  ln_kernel<<<NB, 256, 0, stream>>>(h, g2, b2, ln2, NHID);

  // ffn = gelu(ln2 @ W1^T + b1)  (64x4 = 256 blocks)
  gemm_part_kernel<<<dim3(NFF / 128, 4), 256, 0, stream>>>(ln2, W1, part, NHID, NFF, 4);
  finalize_kernel<<<(NB * NFF) / 256, 256, 0, stream>>>(part, b1, nullptr, ffn, NFF, 4, 1);

  // h2 = ffn @ W2^T + b2m + res2  (K=8192 split 8 ways -> 128 blocks)
  gemm_part_kernel<<<dim3(NHID / 128, 8), 256, 0, stream>>>(ffn, W2, part, NFF, NHID, 8);
  finalize_kernel<<<(NB * NHID) / 256, 256, 0, stream>>>(part, b2m, h, h2, NHID, 8, 0);

  ln_kernel<<<NB, 256, 0, stream>>>(h2, gf, bf, hf, NHID);

  // logits = hf @ Wlm^T  (250 blocks, single pass)
  gemm_kernel<<<NV / 128, 256, 0, stream>>>(hf, Wlm, nullptr, nullptr, out, NHID, NV, 0);
}